// InteractionGNNLayer_61314953118348
// MI455X (gfx1250) — compile-verified
//
#include <hip/hip_runtime.h>
#include <math.h>

typedef float v2f __attribute__((ext_vector_type(2)));
typedef float v8f __attribute__((ext_vector_type(8)));

#define B_     2
#define N_     768
#define D_     256
#define H_     128
#define HD_    16
#define HEADS_ 8
#define NT_    48      // N/16
#define MT_    96      // (B*N)/16
#define RM_    1536    // B*N
#define SPLIT_ 6       // split-K chunks over the key dimension
#define JC_    8       // NT_/SPLIT_ j-tiles per chunk

__device__ __forceinline__ v8f wmma4(v2f a, v2f b, v8f c) {
  // V_WMMA_F32_16X16X4_F32 : D = A(16x4) * B(4x16) + C(16x16), all f32
  return __builtin_amdgcn_wmma_f32_16x16x4_f32(false, a, false, b, (short)0, c, false, false);
}

// ---------------------------------------------------------------------------
// K0: pre-reduce W2 over each head's 16 columns:  W2M[c][h] = mean_d W2[c, h*16+d]
// ---------------------------------------------------------------------------
__global__ void prep_kernel(const float* __restrict__ W2, const float* __restrict__ b2,
                            float* __restrict__ W2M, float* __restrict__ B2M) {
  int t = threadIdx.x;
  if (t < 1024) {
    int c = t >> 3, h = t & 7;
    float s = 0.f;
    #pragma unroll
    for (int d = 0; d < 16; ++d) s += W2[c * H_ + h * 16 + d];
    W2M[c * 8 + h] = s * (1.f / 16.f);
  }
  if (t < 8) {
    float s = 0.f;
    #pragma unroll
    for (int d = 0; d < 16; ++d) s += b2[t * 16 + d];
    B2M[t] = s * (1.f / 16.f);
  }
}

// ---------------------------------------------------------------------------
// K1: Q/K/V projections via f32 WMMA. One 16x16 tile per wave.
// grid: 288 blocks x 256 thr (8 waves) -> 2304 tiles = 96 mtiles * 3 mats * 8 ntiles
// ---------------------------------------------------------------------------
__global__ void __launch_bounds__(256) qkv_kernel(
    const float* __restrict__ X,
    const float* __restrict__ Wq, const float* __restrict__ Wk, const float* __restrict__ Wv,
    float* __restrict__ Q, float* __restrict__ K, float* __restrict__ V) {
  int wave = threadIdx.x >> 5;
  int lane = threadIdx.x & 31;
  int rl = lane & 15, hi = lane >> 4;
  int gid   = blockIdx.x * 8 + wave;
  int mtile = gid / 24;
  int rem   = gid % 24;
  int mat   = rem >> 3;
  int ntile = rem & 7;
  const float* W = (mat == 0) ? Wq : (mat == 1) ? Wk : Wv;
  float*       O = (mat == 0) ? Q  : (mat == 1) ? K  : V;

  const float* arow = X + (mtile * 16 + rl) * D_ + 2 * hi;        // A frag base
  const float* bcol = W + ntile * 16 + rl + 2 * hi * H_;          // B frag base
  v8f c = {};
  #pragma unroll 4
  for (int ks = 0; ks < 64; ++ks) {                               // K = 256 = 64*4
    v2f a = *(const v2f*)(arow + ks * 4);
    v2f b; b[0] = bcol[(ks * 4) * H_]; b[1] = bcol[(ks * 4 + 1) * H_];
    c = wmma4(a, b, c);
  }
  float* orow = O + (mtile * 16 + 8 * hi) * H_ + ntile * 16 + rl;
  #pragma unroll
  for (int r = 0; r < 8; ++r) orow[r * H_] = c[r];
}

// ---------------------------------------------------------------------------
// K2: fused edge-MLP bias + flash attention, split-K over key tiles.
// One block per (split, batch, 16-query tile): 576 blocks of 8 waves.
// Wave h owns head h; all 256 threads cooperate on each 16x16 edge-bias tile.
// Emits per-chunk partial (m, l, unnormalized acc) for the merge kernel.
// ---------------------------------------------------------------------------
__global__ void __launch_bounds__(256) attn_kernel(
    const float* __restrict__ pos, const int* __restrict__ mask,
    const float* __restrict__ Q, const float* __restrict__ K, const float* __restrict__ V,
    const float* __restrict__ W1, const float* __restrict__ b1,
    const float* __restrict__ W2M, const float* __restrict__ B2M,
    float* __restrict__ PACC, float* __restrict__ MBUF, float* __restrict__ LBUF) {
  __shared__ __align__(16) float w1t[128][4];      // {W1[0,c],W1[1,c],W1[2,c],b1[c]}
  __shared__ __align__(16) float w2ms[128][8];
  __shared__ __align__(16) float bias_t[8][16][16];
  __shared__ __align__(16) float pbuf[8][16][18];  // per-head P scratch (even pitch -> b64 ok)

  int sc    = blockIdx.x / (B_ * NT_);
  int rem   = blockIdx.x % (B_ * NT_);
  int b     = rem / NT_;
  int itile = rem % NT_;
  int tid   = threadIdx.x;
  int h     = tid >> 5;
  int lane  = tid & 31;
  int rl = lane & 15, hi = lane >> 4;

  for (int c = tid; c < 128; c += 256) {
    w1t[c][0] = W1[c]; w1t[c][1] = W1[128 + c]; w1t[c][2] = W1[256 + c]; w1t[c][3] = b1[c];
    #pragma unroll
    for (int hh = 0; hh < 8; ++hh) w2ms[c][hh] = W2M[c * 8 + hh];
  }

  int ti = tid >> 4, tj = tid & 15;                 // this thread's edge in the tile
  int gi = itile * 16 + ti;
  float pix = pos[(b * N_ + gi) * 2 + 0];
  float piy = pos[(b * N_ + gi) * 2 + 1];

  const float* qrow = Q + (b * N_ + itile * 16 + rl) * H_ + h * HD_ + 2 * hi;
  v2f qa[4];
  #pragma unroll
  for (int ks = 0; ks < 4; ++ks) qa[ks] = *(const v2f*)(qrow + ks * 4);

  float b2mh[8];
  #pragma unroll
  for (int hh = 0; hh < 8; ++hh) b2mh[hh] = B2M[hh];

  v8f acc = {};
  float mrun[8], lrun[8];
  #pragma unroll
  for (int r = 0; r < 8; ++r) { mrun[r] = -3.0e38f; lrun[r] = 0.f; }

  __syncthreads();

  for (int jj = 0; jj < JC_; ++jj) {
    int jt = sc * JC_ + jj;
    // pull next tile's K/V rows toward the caches while the GELU loop runs
    if (jt + 1 < NT_) {
      __builtin_prefetch(K + (b * N_ + (jt + 1) * 16 + rl) * H_ + h * HD_, 0, 1);
      __builtin_prefetch(V + (b * N_ + (jt + 1) * 16 + rl) * H_ + h * HD_, 0, 1);
    }
    // ---- edge geometry + MLP bias (one edge per thread) ----
    int gj = jt * 16 + tj;
    float pjx = pos[(b * N_ + gj) * 2 + 0];
    float pjy = pos[(b * N_ + gj) * 2 + 1];
    float dx = pix - pjx, dy = piy - pjy;
    float sq   = dx * dx + dy * dy;
    float dist = sqrtf(sq + 1e-6f);
    float dn   = fmaxf(sqrtf(sq), 1e-6f);
    float f0 = dx / dn, f1 = dy / dn, f2 = dist;
    float a8[8];
    #pragma unroll
    for (int hh = 0; hh < 8; ++hh) a8[hh] = b2mh[hh];
    for (int c = 0; c < 128; ++c) {
      float4 w = *(const float4*)&w1t[c][0];
      float x = f0 * w.x + f1 * w.y + f2 * w.z + w.w;
      float g = 0.5f * x * (1.0f + erff(x * 0.70710678118f));   // exact-erf GELU
      const float4* w2p = (const float4*)&w2ms[c][0];
      float4 wa = w2p[0], wb = w2p[1];
      a8[0] += g * wa.x; a8[1] += g * wa.y; a8[2] += g * wa.z; a8[3] += g * wa.w;
      a8[4] += g * wb.x; a8[5] += g * wb.y; a8[6] += g * wb.z; a8[7] += g * wb.w;
    }
    __syncthreads();                       // previous iter's bias_t readers done
    #pragma unroll
    for (int hh = 0; hh < 8; ++hh) bias_t[hh][ti][tj] = a8[hh];
    __syncthreads();                       // bias tile visible

    // ---- scores: S = Q_h(16x16) @ K_h^T(16x16) via 4 WMMA k-steps ----
    v8f s = {};
    const float* krow = K + (b * N_ + jt * 16 + rl) * H_ + h * HD_ + 2 * hi;
    #pragma unroll
    for (int ks = 0; ks < 4; ++ks) {
      v2f kb = *(const v2f*)(krow + ks * 4);
      s = wmma4(qa[ks], kb, s);
    }
    int km = mask[b * N_ + jt * 16 + rl];  // key mask for this lane's column

    // ---- online softmax update (rows live in 16-lane halves) ----
    #pragma unroll
    for (int r = 0; r < 8; ++r) {
      int row = r + 8 * hi;
      float sv = s[r] * 0.25f + bias_t[h][row][rl];   // /SCALE + bias
      sv = km ? sv : -3.0e38f;
      float rmax = sv;
      rmax = fmaxf(rmax, __shfl_xor(rmax, 1, 16));
      rmax = fmaxf(rmax, __shfl_xor(rmax, 2, 16));
      rmax = fmaxf(rmax, __shfl_xor(rmax, 4, 16));
      rmax = fmaxf(rmax, __shfl_xor(rmax, 8, 16));
      float mnew  = fmaxf(mrun[r], rmax);
      float alpha = __expf(mrun[r] - mnew);
      float p     = (sv < -1.0e37f) ? 0.f : __expf(sv - mnew);
      float rsum = p;
      rsum += __shfl_xor(rsum, 1, 16);
      rsum += __shfl_xor(rsum, 2, 16);
      rsum += __shfl_xor(rsum, 4, 16);
      rsum += __shfl_xor(rsum, 8, 16);
      lrun[r] = lrun[r] * alpha + rsum;
      mrun[r] = mnew;
      acc[r]  = acc[r] * alpha;
      pbuf[h][row][rl] = p;                 // C-layout -> LDS for A-layout reload
    }

    // ---- acc += P(16x16) @ V_h(16x16) via 4 WMMA k-steps ----
    const float* vcol = V + (b * N_ + jt * 16 + 2 * hi) * H_ + h * HD_ + rl;
    #pragma unroll
    for (int ks = 0; ks < 4; ++ks) {
      v2f pa = *(const v2f*)&pbuf[h][rl][ks * 4 + 2 * hi];
      v2f vb;
      vb[0] = vcol[(ks * 4) * H_];
      vb[1] = vcol[(ks * 4 + 1) * H_];
      acc = wmma4(pa, vb, acc);
    }
  }

  // ---- store per-chunk partials (unnormalized) ----
  #pragma unroll
  for (int r = 0; r < 8; ++r) {
    int row = r + 8 * hi;
    PACC[((size_t)sc * RM_ + b * N_ + itile * 16 + row) * H_ + h * HD_ + rl] = acc[r];
  }
  if (rl == 0) {                             // lanes 0 and 16 cover rows 0-7 / 8-15
    #pragma unroll
    for (int r = 0; r < 8; ++r) {
      int row = r + 8 * hi;
      int o = ((sc * B_ + b) * HEADS_ + h) * N_ + itile * 16 + row;
      MBUF[o] = mrun[r];
      LBUF[o] = lrun[r];
    }
  }
}

// ---------------------------------------------------------------------------
// K2b: merge split-K partials (log-sum-exp combine) -> normalized AGG.
// grid: RM_*H_/256 = 768 blocks x 256 thr, one output element per thread.
// ---------------------------------------------------------------------------
__global__ void __launch_bounds__(256) combine_kernel(
    const float* __restrict__ PACC, const float* __restrict__ MBUF,
    const float* __restrict__ LBUF, float* __restrict__ AGG) {
  int idx = blockIdx.x * 256 + threadIdx.x;   // < RM_*H_
  int row = idx >> 7;            // /H_
  int col = idx & (H_ - 1);
  int h   = col >> 4;
  int b   = row / N_;
  int i   = row - b * N_;

  float m[SPLIT_];
  float mtot = -3.0e38f;
  #pragma unroll
  for (int s = 0; s < SPLIT_; ++s) {
    m[s] = MBUF[((s * B_ + b) * HEADS_ + h) * N_ + i];
    mtot = fmaxf(mtot, m[s]);
  }
  float ltot = 0.f, a = 0.f;
  #pragma unroll
  for (int s = 0; s < SPLIT_; ++s) {
    float e = __expf(m[s] - mtot);
    ltot += LBUF[((s * B_ + b) * HEADS_ + h) * N_ + i] * e;
    a    += PACC[((size_t)s * RM_ + row) * H_ + col] * e;
  }
  AGG[(size_t)row * H_ + col] = (ltot > 1e-20f) ? a / ltot : 0.f;
}

// ---------------------------------------------------------------------------
// K3: output projection (WMMA, K=128) + mask + residual + LayerNorm, fused.
// One block per 16-row tile; 8 waves x 2 n-tiles cover all 256 output cols.
// ---------------------------------------------------------------------------
__global__ void __launch_bounds__(256) outln_kernel(
    const float* __restrict__ X, const int* __restrict__ mask,
    const float* __restrict__ AGG, const float* __restrict__ Wo, const float* __restrict__ bo,
    const float* __restrict__ gamma, const float* __restrict__ beta,
    float* __restrict__ out) {
  __shared__ __align__(16) float xt[16][260];
  int mtile = blockIdx.x;
  int tid = threadIdx.x;
  int w = tid >> 5, lane = tid & 31, rl = lane & 15, hi = lane >> 4;

  const float* arow = AGG + (mtile * 16 + rl) * H_ + 2 * hi;
  for (int t2 = 0; t2 < 2; ++t2) {
    int nt = w * 2 + t2;
    const float* bcol = Wo + nt * 16 + rl + 2 * hi * D_;
    v8f c = {};
    #pragma unroll 4
    for (int ks = 0; ks < 32; ++ks) {                 // K = 128 = 32*4
      v2f a = *(const v2f*)(arow + ks * 4);
      v2f b; b[0] = bcol[(ks * 4) * D_]; b[1] = bcol[(ks * 4 + 1) * D_];
      c = wmma4(a, b, c);
    }
    #pragma unroll
    for (int r = 0; r < 8; ++r) {
      int row = r + 8 * hi;
      int col = nt * 16 + rl;
      int rg  = mtile * 16 + row;
      float val = c[r] + bo[col];
      xt[row][col] = mask[rg] ? (X[rg * D_ + col] + val) : 0.f;  // residual+mask
    }
  }
  __syncthreads();

  // LayerNorm: 16 threads per row
  int row = tid >> 4, sub = tid & 15;
  float sum = 0.f, sumsq = 0.f;
  #pragma unroll
  for (int k = 0; k < 16; ++k) {
    float x = xt[row][sub + 16 * k];
    sum += x; sumsq += x * x;
  }
  #pragma unroll
  for (int d = 1; d < 16; d <<= 1) {
    sum   += __shfl_xor(sum, d, 16);
    sumsq += __shfl_xor(sumsq, d, 16);
  }
  float mu   = sum * (1.f / 256.f);
  float var  = sumsq * (1.f / 256.f) - mu * mu;
  float rstd = rsqrtf(var + 1e-5f);
  int rg = mtile * 16 + row;
  for (int k = 0; k < 16; ++k) {
    int col = sub + 16 * k;
    out[rg * D_ + col] = (xt[row][col] - mu) * rstd * gamma[col] + beta[col];
  }
}

// ---------------------------------------------------------------------------
extern "C" void kernel_launch(void* const* d_in, const int* in_sizes, int n_in,
                              void* d_out, int out_size, void* d_ws, size_t ws_size,
                              hipStream_t stream) {
  (void)in_sizes; (void)n_in; (void)out_size; (void)ws_size;
  const float* node  = (const float*)d_in[0];
  const float* pos   = (const float*)d_in[1];
  const int*   mask  = (const int*)d_in[2];
  const float* Wq    = (const float*)d_in[3];
  const float* Wk    = (const float*)d_in[4];
  const float* Wv    = (const float*)d_in[5];
  const float* W1    = (const float*)d_in[6];
  const float* b1    = (const float*)d_in[7];
  const float* W2    = (const float*)d_in[8];
  const float* b2    = (const float*)d_in[9];
  const float* Wo    = (const float*)d_in[10];
  const float* bo    = (const float*)d_in[11];
  const float* gamma = (const float*)d_in[12];
  const float* beta  = (const float*)d_in[13];
  float* out = (float*)d_out;

  float* ws   = (float*)d_ws;
  float* Q    = ws;                        // RM_*H_
  float* K    = Q    + RM_ * H_;           // RM_*H_
  float* V    = K    + RM_ * H_;           // RM_*H_
  float* AGG  = V    + RM_ * H_;           // RM_*H_
  float* W2M  = AGG  + RM_ * H_;           // 1024
  float* B2M  = W2M  + 1024;               // 8
  float* PACC = B2M  + 8;                  // SPLIT_*RM_*H_
  float* MBUF = PACC + (size_t)SPLIT_ * RM_ * H_;   // SPLIT_*B_*HEADS_*N_
  float* LBUF = MBUF + SPLIT_ * B_ * HEADS_ * N_;   // SPLIT_*B_*HEADS_*N_

  prep_kernel   <<<1,                 1024, 0, stream>>>(W2, b2, W2M, B2M);
  qkv_kernel    <<<288,                256, 0, stream>>>(node, Wq, Wk, Wv, Q, K, V);
  attn_kernel   <<<SPLIT_ * B_ * NT_,  256, 0, stream>>>(pos, mask, Q, K, V, W1, b1,
                                                         W2M, B2M, PACC, MBUF, LBUF);
  combine_kernel<<<RM_ * H_ / 256,     256, 0, stream>>>(PACC, MBUF, LBUF, AGG);
  outln_kernel  <<<MT_,                256, 0, stream>>>(node, mask, AGG, Wo, bo,
                                                         gamma, beta, out);
}